// RCNNProposal_8400956031315
// MI455X (gfx1250) — compile-verified
//
#include <hip/hip_runtime.h>
#include <hip/hip_bf16.h>
#include <stdint.h>

// ---------------- constants (match reference) ----------------
#define THR_OBJ   0.2f
#define THR_CLS   0.3f
#define MIN_WH    6.16056f
#define MAX_CAND  3000
#define TOP_N     300
#define NMS_THR   0.7f
#define NEG_SCORE (-1e30f)

#define NBINS 4096      // score histogram bins over (0.3, 1.0]
#define NCAND 3008      // MAX_CAND padded to 16
#define NWRD  188       // NCAND/16 16-bit mask words per row
#define CAP   4096      // candidate capacity (>= MAX_CAND + boundary-bin slack)
#define CHUNK 64        // NMS mask rows staged in LDS per chunk (64*188*4 = 48KB)

// ---------------- CDNA5 feature guards ----------------
#if defined(__gfx1250__) && __has_builtin(__builtin_amdgcn_wmma_f32_16x16x4_f32)
#define HAVE_WMMA_X4 1
#else
#define HAVE_WMMA_X4 0
#endif

#if defined(__gfx1250__) && __has_builtin(__builtin_amdgcn_global_load_async_to_lds_b128)
#define HAVE_ASYNC_LDS 1
#else
#define HAVE_ASYNC_LDS 0
#endif

#if HAVE_WMMA_X4
typedef __attribute__((ext_vector_type(2))) float v2f;
typedef __attribute__((ext_vector_type(8))) float v8f;
#endif

#if HAVE_ASYNC_LDS
// Builtin signature (from hipcc diagnostic): param0 is
//   '__attribute__((__vector_size__(4 * sizeof(int)))) int __device__ *'  (int4-vector, AS1)
typedef int v4i_raw __attribute__((vector_size(16)));
typedef __attribute__((address_space(1))) v4i_raw* g_v4i_p;   // global (AS1)
typedef __attribute__((address_space(3))) v4i_raw* l_v4i_p;   // LDS (AS3)
#endif

// 16-byte global->LDS copy: async tensor-data path on gfx1250, plain copy otherwise.
__device__ __forceinline__ void ldscpy16(unsigned* ldst, const unsigned* gsrc) {
#if HAVE_ASYNC_LDS
  __builtin_amdgcn_global_load_async_to_lds_b128(
      (g_v4i_p)(unsigned long long)(uintptr_t)gsrc,
      (l_v4i_p)(unsigned)(uintptr_t)ldst,
      0, 0);
#else
  *(uint4*)ldst = *(const uint4*)gsrc;
#endif
}

__device__ __forceinline__ void ldscpy_wait() {
#if HAVE_ASYNC_LDS
#if __has_builtin(__builtin_amdgcn_s_wait_asynccnt)
  __builtin_amdgcn_s_wait_asynccnt(0);
#else
  asm volatile("s_wait_asynccnt 0x0" ::: "memory");
#endif
#endif
}

// ---------------- helpers ----------------
__device__ __forceinline__ int score_bin(float s) {
  int b = (int)((s - THR_CLS) * ((float)NBINS / 0.7f));
  return b < 0 ? 0 : (b > NBINS - 1 ? NBINS - 1 : b);
}

__device__ __forceinline__ float4 decode_class(int i, int c,
    const float4* __restrict__ bbox4, const float* __restrict__ rois,
    const float* __restrict__ bmean, const float* __restrict__ bstd,
    float W, float H) {
  float bx1 = rois[5 * i + 1], by1 = rois[5 * i + 2];
  float bx2 = rois[5 * i + 3], by2 = rois[5 * i + 4];
  float w = bx2 - bx1 + 1.f, h = by2 - by1 + 1.f;
  float cx = bx1 + 0.5f * (w - 1.f), cy = by1 + 0.5f * (h - 1.f);
  float4 bp = bbox4[(size_t)i * 4 + c];
  float b0 = bp.x * bstd[0] + bmean[0];
  float b1 = bp.y * bstd[1] + bmean[1];
  float b2 = bp.z * bstd[2] + bmean[2];
  float b3 = bp.w * bstd[3] + bmean[3];
  float pcx = b0 * w + cx, pcy = b1 * h + cy;
  float pw = __expf(b2) * w, ph = __expf(b3) * h;
  float x1 = fminf(fmaxf(pcx - 0.5f * (pw - 1.f), 0.f), W - 1.f);
  float y1 = fminf(fmaxf(pcy - 0.5f * (ph - 1.f), 0.f), H - 1.f);
  float x2 = fminf(fmaxf(pcx + 0.5f * (pw - 1.f), 0.f), W - 1.f);
  float y2 = fminf(fmaxf(pcy + 0.5f * (ph - 1.f), 0.f), H - 1.f);
  return make_float4(x1, y1, x2, y2);
}

// ---------------- K0: zero hist + counters + mask (one contiguous region) ----------------
__global__ void k_zero(unsigned* __restrict__ p, int nwords) {
  int i = blockIdx.x * blockDim.x + threadIdx.x;
  if (i < nwords) p[i] = 0u;
}

// ---------------- K1: decode + score + histogram (the 100MB streaming pass) ----------------
__global__ void k_decode_score(const float4* __restrict__ cls4,
                               const float4* __restrict__ bbox4,
                               const float* __restrict__ rois,
                               const float* __restrict__ bmean,
                               const float* __restrict__ bstd,
                               const int* __restrict__ iminfo,
                               float* __restrict__ scores,
                               unsigned* __restrict__ hist, int N) {
  int i = blockIdx.x * blockDim.x + threadIdx.x;
  if (i >= N) return;
  float4 p = cls4[i];
  float m = p.y; int c = 1;
  if (p.z > m) { m = p.z; c = 2; }
  if (p.w > m) { m = p.w; c = 3; }
  float score = NEG_SCORE;
  if (((1.f - p.x) >= THR_OBJ) && (m > THR_CLS)) {
    float H = (float)iminfo[0], W = (float)iminfo[1];
    float4 b = decode_class(i, c, bbox4, rois, bmean, bstd, W, H);
    float bw = b.z - b.x + 1.f, bh = b.w - b.y + 1.f;
    if ((bw >= MIN_WH) || (bh >= MIN_WH)) {
      score = m;
      atomicAdd(&hist[score_bin(m)], 1u);
    }
  }
  scores[i] = score;
}

// ---------------- K2: suffix-scan histogram, find threshold bin ----------------
__global__ __launch_bounds__(1024) void k_threshold(const unsigned* __restrict__ hist,
                                                    int* __restrict__ thrBin) {
  __shared__ unsigned suf[1024];
  int t = threadIdx.x;
  unsigned h0 = hist[4 * t], h1 = hist[4 * t + 1];
  unsigned h2 = hist[4 * t + 2], h3 = hist[4 * t + 3];
  suf[t] = h0 + h1 + h2 + h3;
  __syncthreads();
  for (int off = 1; off < 1024; off <<= 1) {   // reverse Hillis-Steele -> suffix sums
    unsigned v = suf[t] + ((t + off < 1024) ? suf[t + off] : 0u);
    __syncthreads();
    suf[t] = v;
    __syncthreads();
  }
  unsigned run = (t < 1023) ? suf[t + 1] : 0u;
  int best = -1;
  run += h3; if (run >= MAX_CAND) best = 4 * t + 3;
  if (best < 0) { run += h2; if (run >= MAX_CAND) best = 4 * t + 2; }
  if (best < 0) { run += h1; if (run >= MAX_CAND) best = 4 * t + 1; }
  if (best < 0) { run += h0; if (run >= MAX_CAND) best = 4 * t; }
  if (best >= 0) atomicMax(thrBin, best);      // thrBin pre-zeroed; stays 0 if total < MAX_CAND
}

// ---------------- K3: compact candidates above threshold bin ----------------
__global__ void k_compact(const float* __restrict__ scores, const int* __restrict__ thrBin,
                          unsigned* __restrict__ counter,
                          unsigned long long* __restrict__ keys, int N) {
  int i = blockIdx.x * blockDim.x + threadIdx.x;
  if (i >= N) return;
  float s = scores[i];
  if (s > NEG_SCORE * 0.5f) {
    if (score_bin(s) >= *thrBin) {
      unsigned pos = atomicAdd(counter, 1u);
      if (pos < CAP)  // key: score bits desc, then index asc (ties) -> full sort makes order deterministic
        keys[pos] = ((unsigned long long)__float_as_uint(s) << 32) | (unsigned)(~(unsigned)i);
    }
  }
}

// ---------------- K4: bitonic sort (desc) + rebuild candidate boxes ----------------
__global__ __launch_bounds__(1024) void k_sort_build(
    unsigned long long* __restrict__ keys, const unsigned* __restrict__ counter,
    unsigned* __restrict__ candCountC, unsigned* __restrict__ candIdx,
    float4* __restrict__ candBoxes,
    const float4* __restrict__ cls4, const float4* __restrict__ bbox4,
    const float* __restrict__ rois, const float* __restrict__ bmean,
    const float* __restrict__ bstd, const int* __restrict__ iminfo) {
  __shared__ unsigned long long k[CAP];
  int t = threadIdx.x;
  unsigned M = *counter; if (M > CAP) M = CAP;
  for (int x = t; x < CAP; x += 1024) k[x] = (x < (int)M) ? keys[x] : 0ull;
  __syncthreads();
  for (unsigned sz = 2; sz <= CAP; sz <<= 1) {
    for (unsigned j = sz >> 1; j > 0; j >>= 1) {
      for (int x = t; x < CAP; x += 1024) {
        int p = x ^ (int)j;
        if (p > x) {
          unsigned long long va = k[x], vb = k[p];
          bool up = ((x & sz) == 0);
          if (up ? (va < vb) : (va > vb)) { k[x] = vb; k[p] = va; }
        }
      }
      __syncthreads();
    }
  }
  unsigned C = M < MAX_CAND ? M : MAX_CAND;
  if (t == 0) *candCountC = C;
  float H = (float)iminfo[0], W = (float)iminfo[1];
  for (int r = t; r < NCAND; r += 1024) {
    if (r < (int)C) {
      unsigned idx = ~(unsigned)(k[r] & 0xFFFFFFFFull);
      candIdx[r] = idx;
      float4 p = cls4[idx];
      float mm = p.y; int c = 1;
      if (p.z > mm) { mm = p.z; c = 2; }
      if (p.w > mm) { mm = p.w; c = 3; }
      candBoxes[r] = decode_class((int)idx, c, bbox4, rois, bmean, bstd, W, H);
    } else {
      candIdx[r] = 0xFFFFFFFFu;
      candBoxes[r] = make_float4(-1e8f, -1e8f, -1e8f, -1e8f);  // zero-area, never overlaps
    }
  }
}

// ---------------- K5: IoU suppression mask; WMMA computes area[i]+area[j] outer sum ----------------
__global__ __launch_bounds__(32) void k_mask(const float4* __restrict__ cb,
                                             unsigned* __restrict__ mask) {
  int tr = blockIdx.y, tc = blockIdx.x;
  if (tc < tr) return;  // lower triangle stays zero (pre-zeroed)
  int m0 = tr * 16, n0 = tc * 16;
  int lane = threadIdx.x, half = lane >> 4, l15 = lane & 15;
  __shared__ float4 rb[16], cbx[16];
  __shared__ float ar[16], ac[16];
  if (half == 0) {
    float4 b = cb[m0 + lane];
    rb[lane] = b; ar[lane] = (b.z - b.x) * (b.w - b.y);
  } else {
    float4 b = cb[n0 + l15];
    cbx[l15] = b; ac[l15] = (b.z - b.x) * (b.w - b.y);
  }
  __syncthreads();
#if HAVE_WMMA_X4
  // A(16x4): col0 = area_row, col1 = 1 ; B(4x16): row0 = 1, row1 = area_col
  // => C[m][n] = area_row[m] + area_col[n]   (IoU union pre-sum via matrix core)
  v2f a, bf;
  a.x  = half ? 0.f : ar[l15];   // lanes 0-15: K=0; lanes 16-31: K=2 (zero)
  a.y  = half ? 0.f : 1.f;       // K=1 / K=3
  bf.x = half ? 0.f : 1.f;       // B[0][n] / B[2][n]
  bf.y = half ? 0.f : ac[l15];   // B[1][n] / B[3][n]
  v8f cz = {0.f, 0.f, 0.f, 0.f, 0.f, 0.f, 0.f, 0.f};
  v8f sums = __builtin_amdgcn_wmma_f32_16x16x4_f32(false, a, false, bf,
                                                   (short)0, cz, false, false);
#endif
#pragma unroll
  for (int r = 0; r < 8; ++r) {
    int m = half ? (r + 8) : r;  // matches C/D VGPR layout: vgpr r -> rows r (lanes<16) and r+8
    float4 A = rb[m];
    float4 B = cbx[l15];
    float xx1 = fmaxf(A.x, B.x), yy1 = fmaxf(A.y, B.y);
    float xx2 = fminf(A.z, B.z), yy2 = fminf(A.w, B.w);
    float inter = fmaxf(xx2 - xx1, 0.f) * fmaxf(yy2 - yy1, 0.f);
#if HAVE_WMMA_X4
    float asum = sums[r];
#else
    float asum = ar[m] + ac[l15];
#endif
    float uni = fmaxf(asum - inter, 1e-12f);
    bool pred = (inter > NMS_THR * uni) && ((n0 + l15) > (m0 + m));
#if defined(__HIP_DEVICE_COMPILE__)
    unsigned bal = __builtin_amdgcn_ballot_w32(pred);
#else
    unsigned bal = 0;
#endif
    if (lane == 0) {  // ballot layout == C layout: low 16 bits -> row r, high -> row r+8
      mask[(size_t)(m0 + r) * NWRD + tc]     = bal & 0xFFFFu;
      mask[(size_t)(m0 + 8 + r) * NWRD + tc] = bal >> 16;
    }
  }
}

// ---------------- K6: serial greedy NMS, LDS-staged mask chunks, early exit at TOP_N ----------------
__global__ __launch_bounds__(256) void k_nms(const unsigned* __restrict__ mask,
                                             const unsigned* __restrict__ candCountC,
                                             unsigned* __restrict__ keptIdxG,
                                             unsigned* __restrict__ keptCountG) {
  __shared__ unsigned maskL[CHUNK * NWRD];  // 48KB
  __shared__ unsigned keep[NWRD];
  __shared__ unsigned keptIdx[TOP_N];
  __shared__ unsigned nkL;
  int t = threadIdx.x;
  unsigned C = *candCountC;
  if (t < NWRD) {  // init keep bits for valid slots
    unsigned b16 = (unsigned)t * 16u, bits;
    if (C >= b16 + 16u) bits = 0xFFFFu;
    else if (C > b16)   bits = (1u << (C - b16)) - 1u;
    else                bits = 0u;
    keep[t] = bits;
  }
  if (t == 0) nkL = 0;
  __syncthreads();
  bool done = false;
  for (int base = 0; base < (int)C && !done; base += CHUNK) {
    // stage chunk of mask rows into LDS via async global->LDS B128
    for (int q = t; q < (CHUNK * NWRD) / 4; q += 256)
      ldscpy16(&maskL[q * 4], &mask[(size_t)base * NWRD + q * 4]);
    ldscpy_wait();
    __syncthreads();
    int hi = base + CHUNK; if (hi > (int)C) hi = (int)C;
    for (int i = base; i < hi; ++i) {
      unsigned kb = keep[i >> 4];
      bool kk = (kb >> (i & 15)) & 1u;   // uniform
      __syncthreads();                    // reads before updates
      if (kk) {
        if (t < NWRD) keep[t] &= ~maskL[(i - base) * NWRD + t];
        if (t == 0) { unsigned p = nkL; if (p < TOP_N) keptIdx[p] = (unsigned)i; nkL = p + 1; }
      }
      __syncthreads();
      if (nkL >= TOP_N) { done = true; break; }  // only first TOP_N ranks are output
    }
  }
  unsigned nk = nkL < TOP_N ? nkL : TOP_N;
  if (t == 0) *keptCountG = nk;
  for (int r = t; r < (int)nk; r += 256) keptIdxG[r] = keptIdx[r];
}

// ---------------- K7: emit (300,9) output; writes every element (zeros for empty ranks) ----------------
__global__ void k_output(const float* __restrict__ cls,
                         const float4* __restrict__ candBoxes,
                         const unsigned* __restrict__ candIdx,
                         const unsigned* __restrict__ keptIdxG,
                         const unsigned* __restrict__ keptCountG,
                         float* __restrict__ out) {
  int t = blockIdx.x * blockDim.x + threadIdx.x;
  if (t >= TOP_N * 9) return;
  int r = t / 9, c = t % 9;
  float v = 0.f;
  unsigned kc = *keptCountG;
  if ((unsigned)r < kc) {
    unsigned slot = keptIdxG[r];
    if (c >= 1 && c <= 4) {
      float4 b = candBoxes[slot];
      v = (c == 1) ? b.x : (c == 2) ? b.y : (c == 3) ? b.z : b.w;
    } else if (c >= 5) {
      v = cls[(size_t)candIdx[slot] * 4 + (c - 5)];
    }
  }
  out[t] = v;
}

// ---------------- launcher ----------------
extern "C" void kernel_launch(void* const* d_in, const int* in_sizes, int n_in,
                              void* d_out, int out_size, void* d_ws, size_t ws_size,
                              hipStream_t stream) {
  const float4* cls4  = (const float4*)d_in[0];
  const float*  clsf  = (const float*)d_in[0];
  const float4* bbox4 = (const float4*)d_in[1];
  const float*  rois  = (const float*)d_in[2];
  const float*  bmean = (const float*)d_in[3];
  const float*  bstd  = (const float*)d_in[4];
  const int*    iminf = (const int*)d_in[5];
  int N = in_sizes[0] / 4;

  auto al16 = [](size_t x) { return (x + 15) & ~(size_t)15; };
  size_t OFF_HIST = 0;
  size_t OFF_CTR  = OFF_HIST + (size_t)NBINS * 4;                 // 8 u32 counters
  size_t OFF_MASK = OFF_CTR + 32;
  size_t OFF_SCR  = al16(OFF_MASK + (size_t)NCAND * NWRD * 4);    // end of zero region
  size_t ZWORDS   = OFF_SCR / 4;
  size_t OFF_KEYS = al16(OFF_SCR + (size_t)N * 4);
  size_t OFF_CIDX = al16(OFF_KEYS + (size_t)CAP * 8);
  size_t OFF_CBOX = al16(OFF_CIDX + (size_t)NCAND * 4);
  size_t OFF_KEPT = al16(OFF_CBOX + (size_t)NCAND * 16);
  size_t TOTAL    = OFF_KEPT + (size_t)TOP_N * 4;
  if (ws_size < TOTAL) return;

  char* ws = (char*)d_ws;
  unsigned*            hist    = (unsigned*)(ws + OFF_HIST);
  unsigned*            ctr     = (unsigned*)(ws + OFF_CTR);
  int*                 thrBin  = (int*)&ctr[0];
  unsigned*            candCtr = &ctr[1];
  unsigned*            candC   = &ctr[2];
  unsigned*            keptCnt = &ctr[3];
  unsigned*            mask    = (unsigned*)(ws + OFF_MASK);
  float*               scores  = (float*)(ws + OFF_SCR);
  unsigned long long*  keys    = (unsigned long long*)(ws + OFF_KEYS);
  unsigned*            candIdx = (unsigned*)(ws + OFF_CIDX);
  float4*              candBox = (float4*)(ws + OFF_CBOX);
  unsigned*            keptIdx = (unsigned*)(ws + OFF_KEPT);

  k_zero<<<(int)((ZWORDS + 255) / 256), 256, 0, stream>>>((unsigned*)ws, (int)ZWORDS);
  k_decode_score<<<(N + 255) / 256, 256, 0, stream>>>(cls4, bbox4, rois, bmean, bstd,
                                                      iminf, scores, hist, N);
  k_threshold<<<1, 1024, 0, stream>>>(hist, thrBin);
  k_compact<<<(N + 255) / 256, 256, 0, stream>>>(scores, thrBin, candCtr, keys, N);
  k_sort_build<<<1, 1024, 0, stream>>>(keys, candCtr, candC, candIdx, candBox,
                                       cls4, bbox4, rois, bmean, bstd, iminf);
  dim3 gmask(NCAND / 16, NCAND / 16);
  k_mask<<<gmask, 32, 0, stream>>>(candBox, mask);
  k_nms<<<1, 256, 0, stream>>>(mask, candC, keptIdx, keptCnt);
  k_output<<<(TOP_N * 9 + 255) / 256, 256, 0, stream>>>(clsf, candBox, candIdx,
                                                        keptIdx, keptCnt, (float*)d_out);
}